// Conv1d_71743133712692
// MI455X (gfx1250) — compile-verified
//
#include <hip/hip_runtime.h>

typedef float v2f __attribute__((ext_vector_type(2)));
typedef float v8f __attribute__((ext_vector_type(8)));

#define B_SZ   32
#define C_IN   64
#define C_OUT  64
#define L_IN   16384
#define KSZ    3
#define L_OUT  (L_IN - KSZ + 1)   /* 16382 */
#define RDIM   (C_IN * KSZ)       /* 192: weight row length, layout [co][ci][k] */

#define TILE_N 128                /* l-values per block */
#define HALO   (KSZ - 1)          /* 2 */
#define NPAIR  (C_IN / 2)         /* 32 pair-rows */
#define W2     144                /* pair-row stride in float2; 2*W2 mod 64 == 32 -> conflict-free */

// One block = 256 threads = 8 wave32.
//   wave (w&3)  -> co tile   m0 = (w&3)*16
//   wave (w>>2) -> l  base   nb = (w>>2)*64  (four 16-wide n-tiles)
__global__ __launch_bounds__(256)
void conv1d_wmma_f32(const float* __restrict__ x,
                     const float* __restrict__ w,
                     const float* __restrict__ bias,
                     float* __restrict__ out)
{
    // Paired-row im2col stage: tile2[p*W2 + c] = { x[2p][l0+c], x[2p+1][l0+c] }
    __shared__ v2f tile2[NPAIR * W2];   // 36,864 bytes

    const int b   = blockIdx.y;
    const int l0  = blockIdx.x * TILE_N;
    const int tid = threadIdx.x;

    const float* xb = x + (size_t)b * C_IN * L_IN;

    // ---- vectorized fill: columns 0..127 (always in-bounds: l0+127 <= 16383) ----
    #pragma unroll
    for (int it = 0; it < 4; ++it) {
        int i = tid + it * 256;            // 0..1023
        int p = i >> 5;                    // pair row 0..31
        int c = (i & 31) * 4;              // column 0..124, step 4
        const float4 r0 = *(const float4*)(xb + (size_t)(2 * p    ) * L_IN + l0 + c);
        const float4 r1 = *(const float4*)(xb + (size_t)(2 * p + 1) * L_IN + l0 + c);
        v2f* dst = &tile2[p * W2 + c];
        v2f d0; d0.x = r0.x; d0.y = r1.x; dst[0] = d0;
        v2f d1; d1.x = r0.y; d1.y = r1.y; dst[1] = d1;
        v2f d2; d2.x = r0.z; d2.y = r1.z; dst[2] = d2;
        v2f d3; d3.x = r0.w; d3.y = r1.w; dst[3] = d3;
    }
    // ---- halo columns 128,129 (bounds-checked, zero-filled past end of x) ----
    if (tid < 2 * NPAIR) {
        int p  = tid >> 1;
        int c  = TILE_N + (tid & 1);
        int gl = l0 + c;
        v2f d;
        d.x = (gl < L_IN) ? xb[(size_t)(2 * p    ) * L_IN + gl] : 0.0f;
        d.y = (gl < L_IN) ? xb[(size_t)(2 * p + 1) * L_IN + gl] : 0.0f;
        tile2[p * W2 + c] = d;
    }
    __syncthreads();

    const int wave = tid >> 5;
    const int lane = tid & 31;
    const int lo   = lane & 15;   // position within 16-lane half
    const int hi   = lane >> 4;   // which half of the wave

    const int m0 = (wave & 3) * 16;
    const int nb = (wave >> 2) * 64;

    // ---- accumulators, pre-loaded with bias ----
    // C/D layout: VGPR j holds M = m0 + j (lanes 0-15) / m0 + j + 8 (lanes 16-31)
    v8f acc[4];
    #pragma unroll
    for (int j = 0; j < 8; ++j) {
        float bv = bias[m0 + j + 8 * hi];
        acc[0][j] = bv; acc[1][j] = bv; acc[2][j] = bv; acc[3][j] = bv;
    }

    // A-matrix row for this lane: w[co = m0+lo][*][*], 192 contiguous floats
    const float* wrow = w + (size_t)(m0 + lo) * RDIM;

    // Reduce dim ordered r = k*64 + ci, consumed 4 at a time.
    // A 32-bit layout: VGPR0 = K even, VGPR1 = K odd -> lane supplies rows r = cc+2*hi, r+1.
    // B fragment = one aligned ds_load_b64 from the paired-row LDS tile.
    for (int k = 0; k < KSZ; ++k) {
        #pragma unroll 4
        for (int cc = 0; cc < C_IN; cc += 4) {
            const int r = cc + 2 * hi;          // lane's first reduce row (even)
            v2f a;
            a.x = wrow[r * KSZ + k];            // w[m0+lo][r  ][k]
            a.y = wrow[(r + 1) * KSZ + k];      // w[m0+lo][r+1][k]

            const v2f* bp = &tile2[(r >> 1) * W2 + nb + lo + k];
            #pragma unroll
            for (int t = 0; t < 4; ++t) {
                v2f bf = bp[t * 16];            // { B[r][n], B[r+1][n] }
                acc[t] = __builtin_amdgcn_wmma_f32_16x16x4_f32(
                    /*neg_a=*/false, a, /*neg_b=*/false, bf,
                    /*c_mod=*/(short)0, acc[t],
                    /*reuse_a=*/false, /*reuse_b=*/false);
            }
        }
    }

    // ---- store (guard the ragged last l tile) ----
    float* ob = out + (size_t)b * C_OUT * L_OUT;
    const int lbase = l0 + nb + lo;
    #pragma unroll
    for (int t = 0; t < 4; ++t) {
        int l = lbase + t * 16;
        if (l < L_OUT) {
            #pragma unroll
            for (int j = 0; j < 8; ++j) {
                ob[(size_t)(m0 + j + 8 * hi) * L_OUT + l] = acc[t][j];
            }
        }
    }
}

extern "C" void kernel_launch(void* const* d_in, const int* in_sizes, int n_in,
                              void* d_out, int out_size, void* d_ws, size_t ws_size,
                              hipStream_t stream) {
    const float* x    = (const float*)d_in[0];
    const float* w    = (const float*)d_in[1];
    const float* bias = (const float*)d_in[2];
    float* out        = (float*)d_out;

    dim3 grid((L_OUT + TILE_N - 1) / TILE_N, B_SZ);  // (128, 32)
    conv1d_wmma_f32<<<grid, dim3(256, 1, 1), 0, stream>>>(x, w, bias, out);
}